// ScaledDotProductAttention_3642132267477
// MI455X (gfx1250) — compile-verified
//
#include <hip/hip_runtime.h>
#include <math.h>

typedef __attribute__((ext_vector_type(2))) float v2f;
typedef __attribute__((ext_vector_type(4))) float v4f;
typedef __attribute__((ext_vector_type(8))) float v8f;

#define S_LEN 2048
#define SROW  2052   // LDS row stride: 2052 mod 64 banks = 4 -> conflict-free
#define D_DIM 64
#define QT    16     // query rows per block
#define NW    4      // waves per block
#define NTHR  128

__launch_bounds__(NTHR, 2)
__global__ void attn_fwd_f32wmma(const float* __restrict__ q,
                                 const float* __restrict__ k,
                                 const float* __restrict__ v,
                                 float* __restrict__ attn_out,
                                 float* __restrict__ out)
{
    __shared__ float sc[QT * SROW];    // ~128.25 KB score / prob buffer
    __shared__ float red[QT * 8];
    __shared__ float rowmax[QT];
    __shared__ float rowinv[QT];

    const int b    = blockIdx.x >> 7;        // 128 query tiles per batch
    const int qt   = blockIdx.x & 127;
    const int tid  = threadIdx.x;
    const int lane = tid & 31;
    const int wave = tid >> 5;
    const int lr    = lane & 15;             // row (A) / col (B,C) within tile
    const int khalf = lane >> 4;             // 0 or 1
    const int koff  = khalf * 2;             // K sub-offset 0 or 2 (A/B layout)

    const size_t brow = (size_t)b * S_LEN;
    const float* qb = q + brow * D_DIM;
    const float* kb = k + brow * D_DIM;
    const float* vb = v + brow * D_DIM;

    const int ncols = (qt + 1) * 16;         // causal column extent

    // ---- preload Q A-fragments: 16 steps of 16x4 over D=64 ----
    v2f aq[16];
    {
        const float* qrow = qb + (size_t)(qt * 16 + lr) * D_DIM + koff;
        #pragma unroll
        for (int s = 0; s < 16; ++s)
            aq[s] = *(const v2f*)(qrow + 4 * s);
    }

    // ---- stage 1: S = (Q K^T)/8 with causal mask, into LDS ----
    for (int jt = wave; jt <= qt; jt += NW) {
        const float* krow = kb + (size_t)(jt * 16 + lr) * D_DIM + koff;
        v8f acc = {};
        #pragma unroll
        for (int s = 0; s < 16; ++s) {
            v2f bk = *(const v2f*)(krow + 4 * s);
            acc = __builtin_amdgcn_wmma_f32_16x16x4_f32(
                      false, aq[s], false, bk, (short)0, acc, false, false);
        }
        const int n = lr;
        #pragma unroll
        for (int i = 0; i < 8; ++i) {
            int m = i + khalf * 8;
            float val = acc[i] * 0.125f;
            if (jt == qt && n > m) val = -INFINITY;   // causal (future masked)
            sc[m * SROW + jt * 16 + n] = val;
        }
    }
    __syncthreads();

    // ---- softmax over each of 16 rows (first ncols cols valid) ----
    {
        const int r = tid & 15;
        const int c = tid >> 4;                  // 0..7
        float mx = -INFINITY;
        for (int col = c; col < ncols; col += 8)
            mx = fmaxf(mx, sc[r * SROW + col]);
        red[r * 8 + c] = mx;
    }
    __syncthreads();
    if (tid < 16) {
        float mx = red[tid * 8];
        #pragma unroll
        for (int i = 1; i < 8; ++i) mx = fmaxf(mx, red[tid * 8 + i]);
        rowmax[tid] = mx;
    }
    __syncthreads();
    {
        const int r = tid & 15;
        const int c = tid >> 4;
        const float mx = rowmax[r];
        float sm = 0.f;
        for (int col = c; col < ncols; col += 8) {
            float e = __expf(sc[r * SROW + col] - mx);
            sc[r * SROW + col] = e;              // keep unnormalized probs
            sm += e;
        }
        red[r * 8 + c] = sm;
    }
    __syncthreads();
    if (tid < 16) {
        float sm = 0.f;
        #pragma unroll
        for (int i = 0; i < 8; ++i) sm += red[tid * 8 + i];
        rowinv[tid] = 1.0f / sm;
    }
    __syncthreads();

    // ---- write normalized attn to global: float4, non-temporal, coalesced ----
    {
        float* ap = attn_out + (brow + (size_t)qt * 16) * S_LEN;
        for (int pos = tid; pos < QT * (S_LEN / 4); pos += NTHR) {
            int r    = pos >> 9;                 // 512 float4 per row
            int col4 = (pos & 511) << 2;
            v4f val = {0.f, 0.f, 0.f, 0.f};
            if (col4 < ncols) {                  // ncols % 16 == 0 -> uniform over the 4
                v4f e = *(const v4f*)&sc[r * SROW + col4];
                val = e * rowinv[r];
            }
            __builtin_nontemporal_store(val, (v4f*)(ap + (size_t)r * S_LEN + col4));
        }
    }

    // ---- stage 2: out = P @ V ; wave w owns d-cols [16w, 16w+16) ----
    {
        const int dbase = wave * 16;
        const float* vcol = vb + dbase + lr;
        v8f acc = {};
        for (int kk = 0; kk < ncols; kk += 4) {
            v2f a = *(const v2f*)&sc[lr * SROW + kk + koff];    // A: probs (LDS)
            v2f bv;
            bv.x = vcol[(size_t)(kk + koff)     * D_DIM];       // B: V rows
            bv.y = vcol[(size_t)(kk + koff + 1) * D_DIM];
            acc = __builtin_amdgcn_wmma_f32_16x16x4_f32(
                      false, a, false, bv, (short)0, acc, false, false);
        }
        float* op = out + (brow + (size_t)qt * 16) * D_DIM + dbase + lr;
        #pragma unroll
        for (int i = 0; i < 8; ++i) {
            int m = i + khalf * 8;
            __builtin_nontemporal_store(acc[i] * rowinv[m], op + (size_t)m * D_DIM);
        }
    }
}

extern "C" void kernel_launch(void* const* d_in, const int* in_sizes, int n_in,
                              void* d_out, int out_size, void* d_ws, size_t ws_size,
                              hipStream_t stream) {
    (void)in_sizes; (void)n_in; (void)d_ws; (void)ws_size; (void)out_size;
    const float* q = (const float*)d_in[0];
    const float* k = (const float*)d_in[1];
    const float* v = (const float*)d_in[2];
    // d_in[3] is the causal mask; it is reproduced analytically in-kernel.
    float* attn = (float*)d_out;                                  // [16,2048,2048]
    float* outp = attn + (size_t)16 * S_LEN * S_LEN;              // [16,2048,64]
    dim3 grid(16 * (S_LEN / QT));                                 // 2048 blocks
    dim3 block(NTHR);
    attn_fwd_f32wmma<<<grid, block, 0, stream>>>(q, k, v, attn, outp);
}